// GNN_10462540333056
// MI455X (gfx1250) — compile-verified
//
#include <hip/hip_runtime.h>
#include <math.h>

// ---------------------------------------------------------------------------
// 2-layer GCN forward for MI455X (gfx1250, wave32).
//   deg -> dinv -> h0 = x@W1 (WMMA f32 16x16x4) -> edge scatter (f32 atomics,
//   L2-resident) -> relu+bias -> g = h@W2 (WMMA) -> edge scatter -> bias ->
//   column-wise log_softmax (axis=0).
// Edge index streams are tagged non-temporal so the ~46MB node working set
// stays resident in the 192MB L2 (all gathers/atomics hit L2, not HBM).
// ---------------------------------------------------------------------------

typedef __attribute__((ext_vector_type(2))) float v2f;
typedef __attribute__((ext_vector_type(8))) float v8f;

#define F_IN  5
#define F_HID 4
#define F_OUT 7

// ----- ordered-int encoding for float atomicMax ----------------------------
__device__ __forceinline__ int f2oi(float f) {
    int i = __float_as_int(f);
    return (i >= 0) ? i : (i ^ 0x7fffffff);
}
__device__ __forceinline__ float oi2f(int i) {
    return __int_as_float((i >= 0) ? i : (i ^ 0x7fffffff));
}

// ----- init: deg=1 (self loop), accumulators=0, reduction slots ------------
__global__ void k_init(float* deg, float* hacc, float* oacc,
                       int* cmax, float* csum, int n) {
    long long i = (long long)blockIdx.x * blockDim.x + threadIdx.x;
    if (i < n)                 deg[i]  = 1.0f;     // self-loop contributes 1
    if (i < (long long)n * 4)  hacc[i] = 0.0f;
    if (i < (long long)n * 7)  oacc[i] = 0.0f;
    if (i < 7) { cmax[i] = (int)0x80000000; csum[i] = 0.0f; }
}

// ----- degree over dst ------------------------------------------------------
__global__ void k_deg(const int* __restrict__ dst, float* deg, int E) {
    int e = blockIdx.x * blockDim.x + threadIdx.x;
    if (e >= E) return;
    int d = __builtin_nontemporal_load(&dst[e]);
    atomicAdd(&deg[d], 1.0f);
}

__global__ void k_rsqrt(float* dinv, int n) {
    int i = blockIdx.x * blockDim.x + threadIdx.x;
    if (i < n) dinv[i] = rsqrtf(dinv[i]);          // deg >= 1 always
}

// ----- layer-1 GEMM: h0[n,4] = x[n,5] @ W1[5,4]  (V_WMMA_F32_16X16X4_F32) --
// A 16x4 f32 layout: lanes 0-15 -> (K0,K1) in 2 VGPRs; lanes 16-31 -> (K2,K3).
// B 4x16 mirrors: lanes 0-15 -> rows K0,K1; lanes 16-31 -> rows K2,K3.
// D: VGPR v, lanes 0-15 -> M=v,N=lane; lanes 16-31 -> M=v+8,N=lane-16.
__global__ void k_gemm1(const float* __restrict__ x, const float* __restrict__ W1,
                        float* __restrict__ h0, int n) {
    int wave = (blockIdx.x * blockDim.x + threadIdx.x) >> 5;  // 1 wave = 16 rows
    int lane = threadIdx.x & 31;
    int base = wave * 16;
    if (base >= n) return;                                    // wave-uniform
    int half = lane >> 4;
    int l    = lane & 15;
    if (base + 16 > n) {                                      // scalar tail
        int r = base + l;
        if (half == 0 && r < n) {
            #pragma unroll
            for (int j = 0; j < F_HID; ++j) {
                float acc = 0.f;
                #pragma unroll
                for (int k = 0; k < F_IN; ++k)
                    acc = fmaf(x[r * F_IN + k], W1[k * F_HID + j], acc);
                h0[r * F_HID + j] = acc;
            }
        }
        return;
    }
    int row = base + l;
    v2f a0, a1;
    if (half == 0) { a0.x = x[row*F_IN+0]; a0.y = x[row*F_IN+1];
                     a1.x = x[row*F_IN+4]; a1.y = 0.f; }          // K=4 in step 2
    else           { a0.x = x[row*F_IN+2]; a0.y = x[row*F_IN+3];
                     a1.x = 0.f;           a1.y = 0.f; }
    v2f b0 = {0.f, 0.f}, b1 = {0.f, 0.f};
    if (l < F_HID) {
        if (half == 0) { b0.x = W1[0*F_HID+l]; b0.y = W1[1*F_HID+l];
                         b1.x = W1[4*F_HID+l]; }
        else           { b0.x = W1[2*F_HID+l]; b0.y = W1[3*F_HID+l]; }
    }
    v8f c = {};
    c = __builtin_amdgcn_wmma_f32_16x16x4_f32(false, a0, false, b0,
                                              (short)0, c, false, false);
    c = __builtin_amdgcn_wmma_f32_16x16x4_f32(false, a1, false, b1,
                                              (short)0, c, false, false);
    if (l < F_HID) {
        int m0 = base + half * 8;
        #pragma unroll
        for (int v = 0; v < 8; ++v) h0[(m0 + v) * F_HID + l] = c[v];
    }
}

// ----- edge scatter, layer 1 (4 f32 atomics/edge, L2-resident) -------------
__global__ void k_scatter1(const int* __restrict__ src, const int* __restrict__ dst,
                           const float* __restrict__ dinv, const float* __restrict__ h0,
                           float* __restrict__ hacc, int E) {
    int e = blockIdx.x * blockDim.x + threadIdx.x;
    if (e >= E) return;
    int s = __builtin_nontemporal_load(&src[e]);
    int d = __builtin_nontemporal_load(&dst[e]);
    float w = dinv[s] * dinv[d];
    float4 v = ((const float4*)h0)[s];
    atomicAdd(&hacc[d*4+0], w * v.x);
    atomicAdd(&hacc[d*4+1], w * v.y);
    atomicAdd(&hacc[d*4+2], w * v.z);
    atomicAdd(&hacc[d*4+3], w * v.w);
}

// ----- layer 1 finalize: self-loop + bias + relu ---------------------------
__global__ void k_fin1(const float* __restrict__ dinv, const float* __restrict__ h0,
                       const float* __restrict__ hacc, const float* __restrict__ b1,
                       float* __restrict__ h, int n) {
    int i = blockIdx.x * blockDim.x + threadIdx.x;
    if (i >= n) return;
    float w = dinv[i] * dinv[i];
    float4 a = ((const float4*)hacc)[i];
    float4 v = ((const float4*)h0)[i];
    float4 r;
    r.x = fmaxf(fmaf(w, v.x, a.x) + b1[0], 0.f);
    r.y = fmaxf(fmaf(w, v.y, a.y) + b1[1], 0.f);
    r.z = fmaxf(fmaf(w, v.z, a.z) + b1[2], 0.f);
    r.w = fmaxf(fmaf(w, v.w, a.w) + b1[3], 0.f);
    ((float4*)h)[i] = r;
}

// ----- layer-2 GEMM: g[n,7] = h[n,4] @ W2[4,7]  (one WMMA, K=4 exact) ------
__global__ void k_gemm2(const float* __restrict__ h, const float* __restrict__ W2,
                        float* __restrict__ g, int n) {
    int wave = (blockIdx.x * blockDim.x + threadIdx.x) >> 5;
    int lane = threadIdx.x & 31;
    int base = wave * 16;
    if (base >= n) return;
    int half = lane >> 4;
    int l    = lane & 15;
    if (base + 16 > n) {                                      // scalar tail
        int r = base + l;
        if (half == 0 && r < n) {
            #pragma unroll
            for (int j = 0; j < F_OUT; ++j) {
                float acc = 0.f;
                #pragma unroll
                for (int k = 0; k < F_HID; ++k)
                    acc = fmaf(h[r * F_HID + k], W2[k * F_OUT + j], acc);
                g[r * F_OUT + j] = acc;
            }
        }
        return;
    }
    int row = base + l;
    v2f a;
    if (half == 0) { a.x = h[row*F_HID+0]; a.y = h[row*F_HID+1]; }
    else           { a.x = h[row*F_HID+2]; a.y = h[row*F_HID+3]; }
    v2f b = {0.f, 0.f};
    if (l < F_OUT) {
        if (half == 0) { b.x = W2[0*F_OUT+l]; b.y = W2[1*F_OUT+l]; }
        else           { b.x = W2[2*F_OUT+l]; b.y = W2[3*F_OUT+l]; }
    }
    v8f c = {};
    c = __builtin_amdgcn_wmma_f32_16x16x4_f32(false, a, false, b,
                                              (short)0, c, false, false);
    if (l < F_OUT) {
        int m0 = base + half * 8;
        #pragma unroll
        for (int v = 0; v < 8; ++v) g[(m0 + v) * F_OUT + l] = c[v];
    }
}

// ----- edge scatter, layer 2 (7 f32 atomics/edge) --------------------------
__global__ void k_scatter2(const int* __restrict__ src, const int* __restrict__ dst,
                           const float* __restrict__ dinv, const float* __restrict__ g,
                           float* __restrict__ oacc, int E) {
    int e = blockIdx.x * blockDim.x + threadIdx.x;
    if (e >= E) return;
    int s = __builtin_nontemporal_load(&src[e]);
    int d = __builtin_nontemporal_load(&dst[e]);
    float w = dinv[s] * dinv[d];
    const float* gs = g + (size_t)s * F_OUT;
    float* od = oacc + (size_t)d * F_OUT;
    #pragma unroll
    for (int f = 0; f < F_OUT; ++f) atomicAdd(&od[f], w * gs[f]);
}

// ----- layer 2 finalize -> pre-softmax logits in d_out ---------------------
__global__ void k_fin2(const float* __restrict__ dinv, const float* __restrict__ g,
                       const float* __restrict__ oacc, const float* __restrict__ b2,
                       float* __restrict__ out, int n) {
    int i = blockIdx.x * blockDim.x + threadIdx.x;
    if (i >= n) return;
    float w = dinv[i] * dinv[i];
    #pragma unroll
    for (int f = 0; f < F_OUT; ++f)
        out[(size_t)i*F_OUT + f] = fmaf(w, g[(size_t)i*F_OUT + f],
                                        oacc[(size_t)i*F_OUT + f]) + b2[f];
}

// ----- column-wise (axis=0) log-softmax reductions -------------------------
__global__ void k_colmax(const float* __restrict__ out, int* cmax, int n) {
    int tid = blockIdx.x * blockDim.x + threadIdx.x;
    int stride = gridDim.x * blockDim.x;
    float m[F_OUT];
    #pragma unroll
    for (int f = 0; f < F_OUT; ++f) m[f] = -INFINITY;
    for (int i = tid; i < n; i += stride) {
        #pragma unroll
        for (int f = 0; f < F_OUT; ++f)
            m[f] = fmaxf(m[f], out[(size_t)i*F_OUT + f]);
    }
    #pragma unroll
    for (int f = 0; f < F_OUT; ++f) atomicMax(&cmax[f], f2oi(m[f]));
}

__global__ void k_colsum(const float* __restrict__ out, const int* __restrict__ cmax,
                         float* csum, int n) {
    int tid = blockIdx.x * blockDim.x + threadIdx.x;
    int stride = gridDim.x * blockDim.x;
    float m[F_OUT], s[F_OUT];
    #pragma unroll
    for (int f = 0; f < F_OUT; ++f) { m[f] = oi2f(cmax[f]); s[f] = 0.f; }
    for (int i = tid; i < n; i += stride) {
        #pragma unroll
        for (int f = 0; f < F_OUT; ++f)
            s[f] += __expf(out[(size_t)i*F_OUT + f] - m[f]);
    }
    #pragma unroll
    for (int f = 0; f < F_OUT; ++f) atomicAdd(&csum[f], s[f]);
}

__global__ void k_lse(const int* __restrict__ cmax, const float* __restrict__ csum,
                      float* lse) {
    int i = threadIdx.x;
    if (i < F_OUT) lse[i] = oi2f(cmax[i]) + logf(csum[i]);
}

__global__ void k_lsm(float* __restrict__ out, const float* __restrict__ lse,
                      long long total) {
    long long i = (long long)blockIdx.x * blockDim.x + threadIdx.x;
    if (i >= total) return;
    out[i] -= lse[(int)(i % F_OUT)];
}

// ---------------------------------------------------------------------------
extern "C" void kernel_launch(void* const* d_in, const int* in_sizes, int n_in,
                              void* d_out, int out_size, void* d_ws, size_t ws_size,
                              hipStream_t stream) {
    const float* x  = (const float*)d_in[0];
    const int*   ei = (const int*)d_in[1];   // int32: default JAX has x64 disabled
    const float* W1 = (const float*)d_in[2];
    const float* b1 = (const float*)d_in[3];
    const float* W2 = (const float*)d_in[4];
    const float* b2 = (const float*)d_in[5];
    float* out = (float*)d_out;

    const int n = in_sizes[0] / F_IN;
    const int E = in_sizes[1] / 2;
    const int* src = ei;
    const int* dst = ei + E;

    // workspace layout (16B-aligned regions)
    size_t nA = ((size_t)n + 3) & ~(size_t)3;
    float* ws   = (float*)d_ws;
    float* dinv = ws;                 // n   (deg, then rsqrt in place)
    float* h0   = ws + 1  * nA;       // 4n
    float* hacc = ws + 5  * nA;       // 4n
    float* h    = ws + 9  * nA;       // 4n
    float* g    = ws + 13 * nA;       // 7n
    float* oacc = ws + 20 * nA;       // 7n
    int*   cmax = (int*)(ws + 27 * nA);
    float* csum = (float*)(cmax + 8);
    float* lse  = csum + 8;

    const int B = 256;
    auto blocks = [&](long long work) {
        return (unsigned)((work + B - 1) / B);
    };
    long long tiles = ((long long)n + 15) / 16;   // one wave (32 thr) per tile

    k_init    <<<blocks((long long)n * 7), B, 0, stream>>>(dinv, hacc, oacc, cmax, csum, n);
    k_deg     <<<blocks(E), B, 0, stream>>>(dst, dinv, E);
    k_rsqrt   <<<blocks(n), B, 0, stream>>>(dinv, n);
    k_gemm1   <<<blocks(tiles * 32), B, 0, stream>>>(x, W1, h0, n);
    k_scatter1<<<blocks(E), B, 0, stream>>>(src, dst, dinv, h0, hacc, E);
    k_fin1    <<<blocks(n), B, 0, stream>>>(dinv, h0, hacc, b1, h, n);
    k_gemm2   <<<blocks(tiles * 32), B, 0, stream>>>(h, W2, g, n);
    k_scatter2<<<blocks(E), B, 0, stream>>>(src, dst, dinv, g, oacc, E);
    k_fin2    <<<blocks(n), B, 0, stream>>>(dinv, g, oacc, b2, out, n);
    k_colmax  <<<256, B, 0, stream>>>(out, cmax, n);
    k_colsum  <<<256, B, 0, stream>>>(out, cmax, csum, n);
    k_lse     <<<1, 32, 0, stream>>>(cmax, csum, lse);
    k_lsm     <<<blocks((long long)n * F_OUT), B, 0, stream>>>(out, lse, (long long)n * F_OUT);
}